// ChamferLoss_48825188221596
// MI455X (gfx1250) — compile-verified
//
#include <hip/hip_runtime.h>

typedef __attribute__((ext_vector_type(2))) float v2f;
typedef __attribute__((ext_vector_type(8))) float v8f;

#define NN_TILE 2048   // columns staged per LDS super-tile (32 KB of 320 KB)

// ---------------------------------------------------------------------------
// Pack coords [n,3] -> [n,4] = (c0, c1, c2, |c|^2).  Serves as both the
// A-matrix source (-2*c, 1) and the B-matrix source (c, |c|^2) for the
// norm-trick WMMA NN search.
// ---------------------------------------------------------------------------
__global__ void chamfer_pack_kernel(const float* __restrict__ coord,
                                    float* __restrict__ packed, int n) {
  int i = blockIdx.x * blockDim.x + threadIdx.x;
  if (i < n) {
    float c0 = coord[3 * i + 0];
    float c1 = coord[3 * i + 1];
    float c2 = coord[3 * i + 2];
    float4 v;
    v.x = c0; v.y = c1; v.z = c2;
    v.w = c0 * c0 + c1 * c1 + c2 * c2;
    reinterpret_cast<float4*>(packed)[i] = v;
  }
}

// ---------------------------------------------------------------------------
// Fused bidirectional per-batch NN via V_WMMA_F32_16X16X4_F32.
//   blockIdx.y = 0 : rows = pred,  cols = target  -> idx_x
//   blockIdx.y = 1 : rows = target, cols = pred   -> idx_y
// Each block owns 128 rows (8 waves x 16); the matching batch's packed
// columns are staged into LDS once per block (coalesced b128 -> ds_store),
// then every wave sweeps them with ds_load_b64 B-fragments + 4 WMMAs/iter.
// score[i,j] = |y_j|^2 - 2*x_i.y_j  (same argmin as true d2; A[:,3]=1 pairs
// with B[3,:]=|y|^2).
// A layout (f32 16x4): lanes 0-15 -> K=0,1 ; lanes 16-31 -> K=2,3.
// D layout: VGPR v, lanes 0-15 -> M=v ; lanes 16-31 -> M=v+8 ; N=lane%16.
// ---------------------------------------------------------------------------
__global__ __launch_bounds__(256) void chamfer_nn_wmma_kernel(
    const float* __restrict__ pp,   // packed pred   [n,4]
    const float* __restrict__ tp,   // packed target [n,4]
    const int* __restrict__ poff,   // pred cumulative offsets [nb]
    const int* __restrict__ toff,   // target cumulative offsets [nb]
    int nb, int nrows,
    int* __restrict__ idx_x, int* __restrict__ idx_y) {
  const int dir = blockIdx.y;
  const float* __restrict__ rp   = dir ? tp   : pp;    // row-side packed
  const float* __restrict__ cp   = dir ? pp   : tp;    // col-side packed
  const int*   __restrict__ roff = dir ? toff : poff;
  const int*   __restrict__ coff = dir ? poff : toff;
  int*         __restrict__ iout = dir ? idx_y : idx_x;

  const int lane = threadIdx.x & 31;
  const int wave = threadIdx.x >> 5;
  const int row_base = blockIdx.x * 128;        // block-uniform
  const int row0     = row_base + wave * 16;    // wave's 16 rows

  const int  colLane = lane & 15;
  const bool hi      = lane >= 16;
  const int  sel     = hi ? 1 : 0;

  // Block-uniform batch + column range (scalar loop control for free).
  int bb = 0;
  while (bb < nb - 1 && row_base >= roff[bb]) ++bb;
  const int cstart = (bb == 0) ? 0 : coff[bb - 1];
  const int cend   = coff[bb];

  const float2* __restrict__ r2 = reinterpret_cast<const float2*>(rp);

  // A fragment: lanes 0-15 load (x0,x1), lanes 16-31 load (x2,|x|^2).
  int arow = row0 + colLane;
  if (arow >= nrows) arow = nrows - 1;
  if (arow < 0) arow = 0;
  float2 fa = r2[arow * 2 + sel];
  v2f a;
  a.x = -2.0f * fa.x;
  a.y = hi ? 1.0f : -2.0f * fa.y;

  const float BIGV = 3.0e37f;
  float bestv[8];
  int   bestc[8];
#pragma unroll
  for (int v = 0; v < 8; ++v) { bestv[v] = BIGV; bestc[v] = cstart; }

  __shared__ float2 sy[NN_TILE * 2];            // [col][sel], 32 KB
  float4* __restrict__ sy4 = reinterpret_cast<float4*>(sy);
  const float4* __restrict__ gy4 = reinterpret_cast<const float4*>(cp);

  for (int Cs = cstart; Cs < cend; Cs += NN_TILE) {
    const int ncols = (cend - Cs < NN_TILE) ? (cend - Cs) : NN_TILE;

    __syncthreads();  // previous tile fully consumed
    for (int i = threadIdx.x; i < ncols; i += blockDim.x)
      sy4[i] = gy4[Cs + i];                     // b128 load -> ds_store_b128
    __syncthreads();

    // ---- hot loop: 64 columns / iteration, 4 WMMAs fed from LDS ----
    const int nfull = ncols & ~63;
    for (int l0 = 0; l0 < nfull; l0 += 64) {
      const int lc = l0 + colLane;
      float2 f0 = sy[(lc     ) * 2 + sel];      // ds_load_b64
      float2 f1 = sy[(lc + 16) * 2 + sel];
      float2 f2 = sy[(lc + 32) * 2 + sel];
      float2 f3 = sy[(lc + 48) * 2 + sel];
      v2f b0; b0.x = f0.x; b0.y = f0.y;
      v2f b1; b1.x = f1.x; b1.y = f1.y;
      v2f b2; b2.x = f2.x; b2.y = f2.y;
      v2f b3; b3.x = f3.x; b3.y = f3.y;
      v8f z = {0.0f, 0.0f, 0.0f, 0.0f, 0.0f, 0.0f, 0.0f, 0.0f};
      v8f d0 = __builtin_amdgcn_wmma_f32_16x16x4_f32(
          false, a, false, b0, (short)0, z, false, false);
      v8f d1 = __builtin_amdgcn_wmma_f32_16x16x4_f32(
          false, a, false, b1, (short)0, z, false, false);
      v8f d2 = __builtin_amdgcn_wmma_f32_16x16x4_f32(
          false, a, false, b2, (short)0, z, false, false);
      v8f d3 = __builtin_amdgcn_wmma_f32_16x16x4_f32(
          false, a, false, b3, (short)0, z, false, false);

      const int c0 = Cs + lc;
#pragma unroll
      for (int v = 0; v < 8; ++v) {
        // tournament min; strict < keeps lower column on ties
        float m01 = d0[v]; int c01 = c0;
        if (d1[v] < m01) { m01 = d1[v]; c01 = c0 + 16; }
        float m23 = d2[v]; int c23 = c0 + 32;
        if (d3[v] < m23) { m23 = d3[v]; c23 = c0 + 48; }
        if (m23 < m01) { m01 = m23; c01 = c23; }
        if (m01 < bestv[v]) { bestv[v] = m01; bestc[v] = c01; }
      }
    }

    // ---- remainder: branch-free OOB handling (cndmask, EXEC stays all-1s) ----
    for (int l0 = nfull; l0 < ncols; l0 += 16) {
      const int  lc = l0 + colLane;
      const bool ok = lc < ncols;
      const int  cl = ok ? lc : (ncols - 1);
      float2 fb = sy[cl * 2 + sel];
      fb.x = ok ? fb.x : 0.0f;
      fb.y = ok ? fb.y : (hi ? BIGV : 0.0f);   // B[3,:]=BIG -> score=BIG
      v2f bm; bm.x = fb.x; bm.y = fb.y;
      v8f z = {0.0f, 0.0f, 0.0f, 0.0f, 0.0f, 0.0f, 0.0f, 0.0f};
      v8f d = __builtin_amdgcn_wmma_f32_16x16x4_f32(
          false, a, false, bm, (short)0, z, false, false);
      const int col = Cs + lc;
#pragma unroll
      for (int v = 0; v < 8; ++v) {
        float dv = d[v];
        if (dv < bestv[v]) { bestv[v] = dv; bestc[v] = col; }
      }
    }
  }

  // Reduce across the 16 lanes of each half (xor offsets stay within halves).
#pragma unroll
  for (int off = 8; off >= 1; off >>= 1) {
#pragma unroll
    for (int v = 0; v < 8; ++v) {
      float ov = __shfl_xor(bestv[v], off, 32);
      int   oc = __shfl_xor(bestc[v], off, 32);
      if (ov < bestv[v] || (ov == bestv[v] && oc < bestc[v])) {
        bestv[v] = ov; bestc[v] = oc;
      }
    }
  }

  if (colLane == 0) {
    const int base = row0 + (hi ? 8 : 0);
#pragma unroll
    for (int v = 0; v < 8; ++v) {
      if (base + v < nrows) iout[base + v] = bestc[v];
    }
  }
}

// ---------------------------------------------------------------------------
// Per-point loss terms -> per-block partial sums (deterministic tree reduce).
// coord term is pre-divided by the point's segment length so that
// sum == sum_b seg_b/len_b (point_reduction='mean' per batch).
// ---------------------------------------------------------------------------
__global__ __launch_bounds__(256) void chamfer_loss_partial_kernel(
    const float* __restrict__ pc, const float* __restrict__ tc,
    const float* __restrict__ pf, const float* __restrict__ tf,
    const int* __restrict__ poff, const int* __restrict__ toff, int nb,
    const int* __restrict__ idx_x, const int* __restrict__ idx_y,
    int n, int dfeat, float* __restrict__ partials) {
  float coordAcc = 0.0f, featAcc = 0.0f;
  for (int i = blockIdx.x * blockDim.x + threadIdx.x; i < n;
       i += gridDim.x * blockDim.x) {
    int bx = 0; while (bx < nb - 1 && i >= poff[bx]) ++bx;
    int lenx = poff[bx] - (bx ? poff[bx - 1] : 0);
    int by = 0; while (by < nb - 1 && i >= toff[by]) ++by;
    int leny = toff[by] - (by ? toff[by - 1] : 0);
    if (lenx < 1) lenx = 1;
    if (leny < 1) leny = 1;

    const int jx = idx_x[i];   // pred i -> nearest target
    const int jy = idx_y[i];   // target i -> nearest pred
    float dx = 0.0f, dy = 0.0f;
#pragma unroll
    for (int k = 0; k < 3; ++k) {
      float a = pc[3 * i + k] - tc[3 * jx + k];
      float b = tc[3 * i + k] - pc[3 * jy + k];
      dx += a * a;
      dy += b * b;
    }
    coordAcc += dx / (float)lenx + dy / (float)leny;

    float fa = 0.0f;
    for (int f = 0; f < dfeat; ++f) {
      float df = pf[i * dfeat + f] - tf[jx * dfeat + f];
      fa += df * df;
    }
    featAcc += fa;
  }

  __shared__ float sc[256];
  __shared__ float sf[256];
  const int t = threadIdx.x;
  sc[t] = coordAcc; sf[t] = featAcc;
  __syncthreads();
  for (int s = 128; s > 0; s >>= 1) {
    if (t < s) { sc[t] += sc[t + s]; sf[t] += sf[t + s]; }
    __syncthreads();
  }
  if (t == 0) {
    partials[2 * blockIdx.x + 0] = sc[0];
    partials[2 * blockIdx.x + 1] = sf[0];
  }
}

__global__ __launch_bounds__(64) void chamfer_final_kernel(
    const float* __restrict__ partials, int np, int nb, int n, int dfeat,
    float* __restrict__ out) {
  __shared__ float sc[64];
  __shared__ float sf[64];
  const int t = threadIdx.x;
  float c = 0.0f, f = 0.0f;
  for (int i = t; i < np; i += 64) {
    c += partials[2 * i + 0];
    f += partials[2 * i + 1];
  }
  sc[t] = c; sf[t] = f;
  __syncthreads();
  for (int s = 32; s > 0; s >>= 1) {
    if (t < s) { sc[t] += sc[t + s]; sf[t] += sf[t + s]; }
    __syncthreads();
  }
  if (t == 0) {
    float coord_loss = sc[0] / (float)nb;                   // batch mean
    float feat_loss  = sf[0] / ((float)n * (float)dfeat);   // element mean
    float loss = 1.0f * (1.0f * coord_loss + 0.1f * feat_loss);
    out[0] = loss;
    out[1] = coord_loss;
    out[2] = feat_loss;
  }
}

// ---------------------------------------------------------------------------
extern "C" void kernel_launch(void* const* d_in, const int* in_sizes, int n_in,
                              void* d_out, int out_size, void* d_ws,
                              size_t ws_size, hipStream_t stream) {
  const float* pc  = (const float*)d_in[0];  // pred_coord  [n,3]
  const float* tc  = (const float*)d_in[1];  // target_coord[n,3]
  const float* pf  = (const float*)d_in[2];  // pred_feat   [n,d]
  const float* tf  = (const float*)d_in[3];  // target_feat [n,d]
  const int* poff  = (const int*)d_in[4];    // cumulative ends [nb]
  const int* toff  = (const int*)d_in[5];

  const int n     = in_sizes[0] / 3;
  const int nb    = in_sizes[4];
  const int dfeat = (n > 0) ? in_sizes[2] / n : 16;

  char* ws = (char*)d_ws;
  float* xp       = (float*)ws;                                  // n*4 f32
  float* yp       = (float*)(ws + (size_t)n * 16);               // n*4 f32
  int*   idxx     = (int*)  (ws + (size_t)n * 32);               // n i32
  int*   idxy     = (int*)  (ws + (size_t)n * 32 + (size_t)n * 4);
  float* partials = (float*)(ws + (size_t)n * 32 + (size_t)n * 8);

  const int packBlocks = (n + 255) / 256;
  chamfer_pack_kernel<<<packBlocks, 256, 0, stream>>>(pc, xp, n);
  chamfer_pack_kernel<<<packBlocks, 256, 0, stream>>>(tc, yp, n);

  // Fused bidirectional NN: gridDim.y = 2 (pred->target, target->pred).
  const int nnBlocks = (n + 127) / 128;   // 128 rows per block (8 waves x 16)
  chamfer_nn_wmma_kernel<<<dim3(nnBlocks, 2), 256, 0, stream>>>(
      xp, yp, poff, toff, nb, n, idxx, idxy);

  const int rBlocks = 64;
  chamfer_loss_partial_kernel<<<rBlocks, 256, 0, stream>>>(
      pc, tc, pf, tf, poff, toff, nb, idxx, idxy, n, dfeat, partials);
  chamfer_final_kernel<<<1, 64, 0, stream>>>(partials, rBlocks, nb, n, dfeat,
                                             (float*)d_out);
}